// GCN_60954175865280
// MI455X (gfx1250) — compile-verified
//
#include <hip/hip_runtime.h>

// ---------------------------------------------------------------------------
// 3-layer GCN for MI455X (gfx1250, wave32).
//   layer: H = act(X) @ W ; out = bias + dinv^2 * H (self loop)
//          out[dst] += dinv[src]*dinv[dst] * H[src]  (edge scatter, atomics)
// GEMM uses V_WMMA_F32_16X16X4_F32 (fp32 in/out, matches reference numerics).
// ---------------------------------------------------------------------------

typedef __attribute__((ext_vector_type(2))) float v2f;
typedef __attribute__((ext_vector_type(8))) float v8f;

#define N_NODES 100000
#define E_EDGES 1600000
#define K_DIM   128   // inner dim is 128 for all three layers

// ---------------------------------------------------------------- degree prep
__global__ __launch_bounds__(256) void deg_init(float* __restrict__ deg) {
    int i = blockIdx.x * 256 + threadIdx.x;
    if (i < N_NODES) deg[i] = 1.0f;               // self-loop contributes 1
}

__global__ __launch_bounds__(256) void deg_count(const long long* __restrict__ dst,
                                                 float* __restrict__ deg) {
    int e = blockIdx.x * 256 + threadIdx.x;
    if (e < E_EDGES) atomicAdd(&deg[(int)dst[e]], 1.0f);
}

__global__ __launch_bounds__(256) void deg_rsqrt(float* __restrict__ deg) {
    int i = blockIdx.x * 256 + threadIdx.x;
    if (i < N_NODES) deg[i] = rsqrtf(deg[i]);     // deg >= 1 always
}

// ------------------------------------------------------------ WMMA fp32 GEMM
// H[N x COUT] = act(X[N x 128]) @ W[128 x COUT]
// One wave computes one 16x16 tile of H; COUT/16 waves per block cover 16 rows.
template <int COUT, bool RELU>
__global__ __launch_bounds__(32 * (COUT / 16)) void gemm_wmma(
    const float* __restrict__ X, const float* __restrict__ W,
    float* __restrict__ H) {
    static_assert(COUT % 16 == 0, "");
    const int lane = threadIdx.x & 31;
    const int wcol = (threadIdx.x >> 5) * 16;      // this wave's column tile
    const int col  = wcol + (lane & 15);
    const int hi   = lane >> 4;                    // lane half: K sub-pair
    const long tile = blockIdx.x;                  // 16-row tile (N%16==0)
    const long row  = tile * 16 + (lane & 15);

    // Hoist all 32 B fragments (K=128 in steps of 4). W is tiny -> L2/L0 hot.
    v2f b[32];
#pragma unroll
    for (int s = 0; s < 32; ++s) {
        const int k = 4 * s + 2 * hi;
        b[s].x = W[(long)k * COUT + col];
        b[s].y = W[(long)(k + 1) * COUT + col];
    }

    const float* xrow = X + row * K_DIM;
    v8f c = {};
#pragma unroll
    for (int s = 0; s < 32; ++s) {
        const int k = 4 * s + 2 * hi;
        v2f a;
        a.x = xrow[k];
        a.y = xrow[k + 1];
        if (RELU) {                                // fold ReLU of prev layer
            a.x = fmaxf(a.x, 0.0f);
            a.y = fmaxf(a.y, 0.0f);
        }
        // D = A(16x4) * B(4x16) + C   (fp32 WMMA)
        c = __builtin_amdgcn_wmma_f32_16x16x4_f32(
                /*neg_a=*/false, a, /*neg_b=*/false, b[s],
                /*c_mod=*/(short)0, c, /*reuse_a=*/false, /*reuse_b=*/false);
    }

    // D layout: VGPR v <-> row (v + 8*hi), lane&15 <-> col
#pragma unroll
    for (int v = 0; v < 8; ++v) {
        const long m = tile * 16 + v + hi * 8;
        H[m * COUT + col] = c[v];
    }
}

// ------------------------------------------------- out = bias + dinv^2 * H
template <int COUT>
__global__ __launch_bounds__(256) void init_out(
    const float* __restrict__ H, const float* __restrict__ bias,
    const float* __restrict__ dinv, float* __restrict__ Out) {
    const long total = (long)N_NODES * COUT / 4;
    long idx = (long)blockIdx.x * 256 + threadIdx.x;
    if (idx >= total) return;
    const long e0   = idx * 4;
    const long node = e0 / COUT;
    const int  c0   = (int)(e0 % COUT);
    float w = dinv[node];
    w = w * w;
    const float4 hv = *(const float4*)(H + e0);
    const float4 bv = *(const float4*)(bias + c0);
    float4 o;
    o.x = bv.x + hv.x * w;
    o.y = bv.y + hv.y * w;
    o.z = bv.z + hv.z * w;
    o.w = bv.w + hv.w * w;
    *(float4*)(Out + e0) = o;
}

// ------------------------------------- edge scatter: one wave32 per edge
template <int COUT>
__global__ __launch_bounds__(256) void scatter_add(
    const float* __restrict__ H, const long long* __restrict__ src,
    const long long* __restrict__ dst, const float* __restrict__ dinv,
    float* __restrict__ Out) {
    static_assert(COUT == 128 || COUT == 64, "");
    const int  lane = threadIdx.x & 31;
    const long e    = (long)blockIdx.x * 8 + (threadIdx.x >> 5);
    if (e >= E_EDGES) return;
    const int   s = (int)src[e];
    const int   d = (int)dst[e];
    const float w = dinv[s] * dinv[d];
    constexpr int V = COUT / 32;                   // 4 or 2 floats per lane
    const float* hp = H + (long)s * COUT + lane * V;
    float*       op = Out + (long)d * COUT + lane * V;
    if (V == 4) {
        const float4 v = *(const float4*)hp;
        atomicAdd(op + 0, v.x * w);
        atomicAdd(op + 1, v.y * w);
        atomicAdd(op + 2, v.z * w);
        atomicAdd(op + 3, v.w * w);
    } else {
        const float2 v = *(const float2*)hp;
        atomicAdd(op + 0, v.x * w);
        atomicAdd(op + 1, v.y * w);
    }
}

// ---------------------------------------------------------------------------
extern "C" void kernel_launch(void* const* d_in, const int* in_sizes, int n_in,
                              void* d_out, int out_size, void* d_ws, size_t ws_size,
                              hipStream_t stream) {
    const float*     x   = (const float*)d_in[0];
    const long long* ei  = (const long long*)d_in[1];   // int64 (2, E)
    const long long* src = ei;
    const long long* dst = ei + E_EDGES;
    const float* W1 = (const float*)d_in[2];
    const float* b1 = (const float*)d_in[3];
    const float* W2 = (const float*)d_in[4];
    const float* b2 = (const float*)d_in[5];
    const float* W3 = (const float*)d_in[6];
    const float* b3 = (const float*)d_in[7];

    float* A    = (float*)d_ws;                         // N x 128 scratch
    float* B    = A + (long)N_NODES * 128;              // N x 128 scratch
    float* dinv = B + (long)N_NODES * 128;              // N floats
    float* out  = (float*)d_out;                        // N x 64

    const int nblkN = (N_NODES + 255) / 256;
    const int nblkE = (E_EDGES + 255) / 256;
    const int tiles = N_NODES / 16;                     // 6250, exact
    const int eWave = (E_EDGES + 7) / 8;                // 8 edges per block
    const int i128  = (N_NODES * 128 / 4 + 255) / 256;
    const int i64   = (N_NODES * 64 / 4 + 255) / 256;

    // graph normalization: dinv = rsqrt(1 + in-degree)
    deg_init<<<nblkN, 256, 0, stream>>>(dinv);
    deg_count<<<nblkE, 256, 0, stream>>>(dst, dinv);
    deg_rsqrt<<<nblkN, 256, 0, stream>>>(dinv);

    // ---- layer 1: h1 = x @ W1 ; out1 = b1 + D^-1/2 A D^-1/2 h1  -> B
    gemm_wmma<128, false><<<tiles, 256, 0, stream>>>(x, W1, A);
    init_out<128><<<i128, 256, 0, stream>>>(A, b1, dinv, B);
    scatter_add<128><<<eWave, 256, 0, stream>>>(A, src, dst, dinv, B);

    // ---- layer 2: h2 = relu(out1) @ W2 -> A ; out2 -> B (out1 consumed)
    gemm_wmma<128, true><<<tiles, 256, 0, stream>>>(B, W2, A);
    init_out<128><<<i128, 256, 0, stream>>>(A, b2, dinv, B);
    scatter_add<128><<<eWave, 256, 0, stream>>>(A, src, dst, dinv, B);

    // ---- layer 3: h3 = relu(out2) @ W3 -> A (N x 64) ; out -> d_out
    gemm_wmma<64, true><<<tiles, 128, 0, stream>>>(B, W3, A);
    init_out<64><<<i64, 256, 0, stream>>>(A, b3, dinv, out);
    scatter_add<64><<<eWave, 256, 0, stream>>>(A, src, dst, dinv, out);
}